// GCN_9998683865211
// MI455X (gfx1250) — compile-verified
//
#include <hip/hip_runtime.h>

typedef __attribute__((ext_vector_type(2))) float v2f;
typedef __attribute__((ext_vector_type(8))) float v8f;

#define NNODES 100000
#define NEDGES 640000
#define NGRAPH 64
#define LEAKY  0.01f
#define BNEPS  1e-5f

typedef __attribute__((address_space(1))) int* gptr_i32;
typedef __attribute__((address_space(3))) int* lptr_i32;

__device__ __forceinline__ void atomAddF(float* p, float v) {
    // hardware global_atomic_add_f32 (no CAS loop)
    unsafeAtomicAdd(p, v);
}

// ---------------- graph normalization ----------------

__global__ void k_deg(const int* __restrict__ src, const int* __restrict__ dst,
                      float* __restrict__ deg) {
    int e = blockIdx.x * blockDim.x + threadIdx.x;
    if (e < NEDGES) {
        int s = src[e];
        if (s != dst[e]) atomAddF(&deg[s], 1.0f);
    }
}

__global__ void k_dinv(float* __restrict__ deg) {
    int i = blockIdx.x * blockDim.x + threadIdx.x;
    if (i < NNODES) {
        float d = deg[i];
        deg[i] = (d > 0.0f) ? rsqrtf(fmaxf(d, 1.0f)) : 0.0f;
    }
}

__global__ void k_edgew(const int* __restrict__ src, const int* __restrict__ dst,
                        const float* __restrict__ dinv, float* __restrict__ w) {
    int e = blockIdx.x * blockDim.x + threadIdx.x;
    if (e < NEDGES) {
        int s = src[e], d = dst[e];
        w[e] = (s != d) ? (-dinv[s] * dinv[d]) : 0.0f;
    }
}

// out[i] = alpha * in[i]
__global__ void k_scalecopy(const float* __restrict__ in, float* __restrict__ out,
                            float alpha, int n) {
    int i = blockIdx.x * blockDim.x + threadIdx.x;
    if (i < n) out[i] = alpha * in[i];
}

// out[dst[e], :] += alpha * w[e] * X[src[e], :]   (4 features per thread)
__global__ void k_scatter(const int* __restrict__ src, const int* __restrict__ dst,
                          const float* __restrict__ w, const float* __restrict__ X,
                          float* __restrict__ out, int F4, float alpha, int total) {
    int id = blockIdx.x * blockDim.x + threadIdx.x;
    if (id >= total) return;
    int e = id / F4;
    int c = id - e * F4;
    float wv = alpha * w[e];
    if (wv == 0.0f) return;
    const float4 xv = reinterpret_cast<const float4*>(X + (long)src[e] * (F4 * 4))[c];
    float* o = out + (long)dst[e] * (F4 * 4) + c * 4;
    atomAddF(o + 0, wv * xv.x);
    atomAddF(o + 1, wv * xv.y);
    atomAddF(o + 2, wv * xv.z);
    atomAddF(o + 3, wv * xv.w);
}

// ---------------- fused ChebConv GEMM + bias + BN + leaky ReLU ----------------
// One wave computes a full 16-row x FOUT strip (NT=FOUT/16 accumulators) via
// V_WMMA_F32_16X16X4_F32; weights staged per Chebyshev slice in LDS.
template <int FIN, int FOUT>
__global__ void k_cheb_gemm_bn(const float* __restrict__ T0, const float* __restrict__ T1,
                               const float* __restrict__ T2, const float* __restrict__ W,
                               const float* __restrict__ bias, const float* __restrict__ gamma,
                               const float* __restrict__ beta, const float* __restrict__ mean,
                               const float* __restrict__ var, float* __restrict__ out,
                               int nStrips) {
    constexpr int NT = FOUT / 16;
    __shared__ float sW[FIN * FOUT];

    int wave = (blockIdx.x * blockDim.x + threadIdx.x) >> 5;
    int lane = threadIdx.x & 31;
    bool active = wave < nStrips;
    int mt = active ? wave : 0;            // inactive waves compute a redundant strip, never store

    int lrow = lane & 15;                  // M row (A) / N col (B, C/D)
    int kgrp = (lane >> 4) << 1;           // lanes 0-15 -> K{0,1}; lanes 16-31 -> K{2,3}

    v8f zero = {};
    v8f acc[NT];
#pragma unroll
    for (int t = 0; t < NT; ++t) acc[t] = zero;

    const long rowOff = (long)(mt * 16 + lrow) * FIN;

    for (int cb = 0; cb < 3; ++cb) {
        const float* T  = (cb == 0) ? T0 : ((cb == 1) ? T1 : T2);
        const float* Wc = W + cb * FIN * FOUT;

        __syncthreads();                   // protect previous slice's LDS reads
#if __has_builtin(__builtin_amdgcn_global_load_async_to_lds_b32)
        for (int i = threadIdx.x; i < FIN * FOUT; i += 256) {
            __builtin_amdgcn_global_load_async_to_lds_b32(
                (gptr_i32)(unsigned long long)(const char*)(Wc + i),
                (lptr_i32)(unsigned int)(unsigned long long)(char*)(sW + i),
                0, 0);
        }
#if __has_builtin(__builtin_amdgcn_s_wait_asynccnt)
        __builtin_amdgcn_s_wait_asynccnt(0);
#else
        asm volatile("s_wait_asynccnt 0" ::: "memory");
#endif
#else
        for (int i = threadIdx.x; i < FIN * FOUT; i += 256) sW[i] = Wc[i];
#endif
        __syncthreads();

        const float* arow = T + rowOff;
        for (int kk = 0; kk < FIN; kk += 4) {
            __builtin_prefetch(arow + kk + 32, 0, 0);
            v2f a = *reinterpret_cast<const v2f*>(arow + kk + kgrp);  // A[M=lrow, K=kk+kgrp..+1]
#pragma unroll
            for (int t = 0; t < NT; ++t) {
                v2f b;
                b.x = sW[(kk + kgrp) * FOUT + t * 16 + lrow];          // B[K, N]
                b.y = sW[(kk + kgrp + 1) * FOUT + t * 16 + lrow];
                acc[t] = __builtin_amdgcn_wmma_f32_16x16x4_f32(
                    false, a, false, b, (short)0, acc[t], false, false);
            }
        }
    }

    if (!active) return;

    int rowBase = mt * 16 + ((lane >> 4) << 3);  // C/D: VGPR j -> rows j (lanes 0-15), j+8 (16-31)
#pragma unroll
    for (int t = 0; t < NT; ++t) {
        int col = t * 16 + lrow;
        float s   = rsqrtf(var[col] + BNEPS) * gamma[col];
        float off = bias[col] - mean[col];
        float bb  = beta[col];
#pragma unroll
        for (int j = 0; j < 8; ++j) {
            float val = (acc[t][j] + off) * s + bb;
            val = (val > 0.0f) ? val : LEAKY * val;
            out[(long)(rowBase + j) * FOUT + col] = val;
        }
    }
}

// ---------------- pooling + head ----------------

__global__ void k_pool(const float* __restrict__ h, const int* __restrict__ batch,
                       float* __restrict__ sums, float* __restrict__ cnt) {
    int id = blockIdx.x * blockDim.x + threadIdx.x;
    if (id >= NNODES * 16) return;
    int i = id >> 4, f = id & 15;
    int b = batch[i];
    atomAddF(&sums[b * 16 + f], h[id]);
    if (f == 0) atomAddF(&cnt[b], 1.0f);
}

__global__ void k_head(const float* __restrict__ sums, const float* __restrict__ cnt,
                       const float* __restrict__ lw, const float* __restrict__ lb,
                       float* __restrict__ out) {
    int id = threadIdx.x;
    if (id >= NGRAPH * 2) return;
    int g = id >> 1, c = id & 1;
    float inv = 1.0f / fmaxf(cnt[g], 1.0f);
    float acc = lb[c];
#pragma unroll
    for (int f = 0; f < 16; ++f) acc += sums[g * 16 + f] * inv * lw[c * 16 + f];
    out[id] = acc;
}

// ---------------- host side ----------------

template <int FIN, int FOUT>
static void run_layer(const float* T0,
                      const float* W, const float* bias, const float* g, const float* be,
                      const float* m, const float* v,
                      const int* src, const int* dst, const float* w,
                      float* A, float* B, float* out, hipStream_t stream) {
    int F4 = FIN / 4;
    int total = NEDGES * F4;
    size_t nf = (size_t)NNODES * FIN;

    // Tx1 = prop(T0)
    (void)hipMemsetAsync(A, 0, nf * sizeof(float), stream);
    k_scatter<<<(total + 255) / 256, 256, 0, stream>>>(src, dst, w, T0, A, F4, 1.0f, total);
    // Tx2 = 2*prop(Tx1) - T0   (init B = -T0, scatter with alpha=2)
    k_scalecopy<<<((int)nf + 255) / 256, 256, 0, stream>>>(T0, B, -1.0f, (int)nf);
    k_scatter<<<(total + 255) / 256, 256, 0, stream>>>(src, dst, w, A, B, F4, 2.0f, total);
    // out = leaky(BN(T0@W0 + Tx1@W1 + Tx2@W2 + bias))
    int nStrips = NNODES / 16;                       // 6250, one wave per 16-row strip
    int blocks = (nStrips * 32 + 255) / 256;
    k_cheb_gemm_bn<FIN, FOUT><<<blocks, 256, 0, stream>>>(
        T0, A, B, W, bias, g, be, m, v, out, nStrips);
}

extern "C" void kernel_launch(void* const* d_in, const int* in_sizes, int n_in,
                              void* d_out, int out_size, void* d_ws, size_t ws_size,
                              hipStream_t stream) {
    const float* x    = (const float*)d_in[0];
    const int*   ei   = (const int*)d_in[1];
    const int*   src  = ei;
    const int*   dst  = ei + NEDGES;
    const int*   batch = (const int*)d_in[2];
    const float *W1 = (const float*)d_in[3],  *b1 = (const float*)d_in[4],
                *g1 = (const float*)d_in[5],  *be1 = (const float*)d_in[6],
                *m1 = (const float*)d_in[7],  *v1 = (const float*)d_in[8];
    const float *W2 = (const float*)d_in[9],  *b2 = (const float*)d_in[10],
                *g2 = (const float*)d_in[11], *be2 = (const float*)d_in[12],
                *m2 = (const float*)d_in[13], *v2 = (const float*)d_in[14];
    const float *W3 = (const float*)d_in[15], *b3 = (const float*)d_in[16],
                *g3 = (const float*)d_in[17], *be3 = (const float*)d_in[18],
                *m3 = (const float*)d_in[19], *v3 = (const float*)d_in[20];
    const float *lw = (const float*)d_in[21], *lb = (const float*)d_in[22];

    float* ws = (float*)d_ws;
    float* A    = ws;                              // N*128
    float* B    = A + (size_t)NNODES * 128;        // N*128
    float* C    = B + (size_t)NNODES * 128;        // N*64   (h1, later h3)
    float* D    = C + (size_t)NNODES * 64;         // N*32   (h2)
    float* deg  = D + (size_t)NNODES * 32;         // N      (becomes dinv)
    float* w    = deg + NNODES;                    // E
    float* sums = w + NEDGES;                      // G*16
    float* cnt  = sums + NGRAPH * 16;              // G

    // edge weights: w[e] = -dinv[src]*dinv[dst]
    (void)hipMemsetAsync(deg, 0, NNODES * sizeof(float), stream);
    k_deg<<<(NEDGES + 255) / 256, 256, 0, stream>>>(src, dst, deg);
    k_dinv<<<(NNODES + 255) / 256, 256, 0, stream>>>(deg);
    k_edgew<<<(NEDGES + 255) / 256, 256, 0, stream>>>(src, dst, deg, w);

    // three ChebConv+BN+leaky layers
    run_layer<128, 64>(x, W1, b1, g1, be1, m1, v1, src, dst, w, A, B, C, stream);  // h1 -> C
    run_layer< 64, 32>(C, W2, b2, g2, be2, m2, v2, src, dst, w, A, B, D, stream);  // h2 -> D
    run_layer< 32, 16>(D, W3, b3, g3, be3, m3, v3, src, dst, w, A, B, C, stream);  // h3 -> C

    // global mean pool + linear head
    (void)hipMemsetAsync(sums, 0, (NGRAPH * 16 + NGRAPH) * sizeof(float), stream);
    k_pool<<<(NNODES * 16 + 255) / 256, 256, 0, stream>>>(C, batch, sums, cnt);
    k_head<<<1, 128, 0, stream>>>(sums, cnt, lw, lb, (float*)d_out);
}